// GCN_69844758167859
// MI455X (gfx1250) — compile-verified
//
#include <hip/hip_runtime.h>
#include <hip/hip_bf16.h>
#include <math.h>

// ---------------------------------------------------------------------------
// GCN inference on MI455X (gfx1250, wave32).
// Dimensions fixed by the reference:
#define NNODES 65536
#define NEDGES 1048576
#define INDIM  128
#define HID    64
#define ODIM   40
#define OPAD   48   // ODIM padded to a multiple of 16 for WMMA tiles
// ---------------------------------------------------------------------------

typedef __attribute__((ext_vector_type(2))) float v2f;
typedef __attribute__((ext_vector_type(8))) float v8f;

// ---------------- degree / norm ----------------
__global__ void k_init_deg(float* deg) {
    int i = blockIdx.x * blockDim.x + threadIdx.x;
    if (i < NNODES) deg[i] = 1.0f;            // self-loop fill weight = 1
}

__global__ void k_accum_deg(const int* __restrict__ col,
                            const float* __restrict__ w,
                            float* __restrict__ deg) {
    int e = blockIdx.x * blockDim.x + threadIdx.x;
    if (e < NEDGES) atomicAdd(&deg[col[e]], w[e]);
}

__global__ void k_finalize_dis(float* deg) {
    int i = blockIdx.x * blockDim.x + threadIdx.x;
    if (i < NNODES) {
        float d = deg[i];
        deg[i] = (d > 0.0f) ? rsqrtf(d) : 0.0f;   // deg buffer becomes dis
    }
}

// ---------------- fp32 WMMA GEMM: C[M,N] = A[M,K] * B[K,N] ----------------
// Row-major. M,N multiples of 16; K multiple of 4. One 16x16 tile per wave,
// 4 waves per block. Full-EXEC waves as required by WMMA.
__global__ void __launch_bounds__(128)
k_gemm_f32(const float* __restrict__ A, const float* __restrict__ B,
           float* __restrict__ C, int M, int K, int N) {
    const int lane = threadIdx.x;                 // 0..31
    const int tile = blockIdx.x * 4 + threadIdx.y;
    const int ntn  = N >> 4;
    const int m0   = (tile / ntn) << 4;
    const int n0   = (tile % ntn) << 4;
    const int lm   = lane & 15;
    const int kh   = (lane >> 4) << 1;            // 0 for lanes 0-15, 2 for 16-31

    const float* __restrict__ arow = A + (size_t)(m0 + lm) * K;
    const float* __restrict__ bcol = B + n0 + lm;

    v8f acc = {};
    for (int k = 0; k < K; k += 4) {
        v2f a, b;
        // A 16x4 f32 layout: VGPR0 -> K = k+kh, VGPR1 -> K = k+kh+1
        a.x = arow[k + kh];
        a.y = arow[k + kh + 1];
        // B 4x16 f32 layout (rows striped across lanes, symmetric to A)
        b.x = bcol[(size_t)(k + kh)     * N];
        b.y = bcol[(size_t)(k + kh + 1) * N];
        acc = __builtin_amdgcn_wmma_f32_16x16x4_f32(
                  /*neg_a=*/false, a, /*neg_b=*/false, b,
                  /*c_mod=*/(short)0, acc, /*reuse_a=*/false, /*reuse_b=*/false);
    }

    // C/D 16x16 f32 layout: VGPR i -> row m0+i (lanes 0-15) / m0+8+i (lanes 16-31)
    const int mbase = m0 + ((lane >> 4) << 3);
    float* __restrict__ crow = C + n0 + lm;
#pragma unroll
    for (int i = 0; i < 8; ++i)
        crow[(size_t)(mbase + i) * N] = acc[i];
}

// ---------------- aggregation ----------------
// agg[i,:] = dis[i]^2 * ht[i,:]   (self-loop term, also initializes agg)
__global__ void k_self_init(const float* __restrict__ ht,
                            const float* __restrict__ dis,
                            float* __restrict__ agg) {
    int t = blockIdx.x * blockDim.x + threadIdx.x;   // NNODES*HID threads
    int i = t >> 6;
    float d = dis[i];
    agg[t] = d * d * ht[t];
}

// agg[col,:] += dis[row]*w*dis[col] * ht[row,:]  -- one (edge,dim) per thread
__global__ void k_edge_scatter(const int* __restrict__ row,
                               const int* __restrict__ col,
                               const float* __restrict__ w,
                               const float* __restrict__ dis,
                               const float* __restrict__ ht,
                               float* __restrict__ agg) {
    int t = blockIdx.x * blockDim.x + threadIdx.x;   // NEDGES*HID threads
    int e = t >> 6;
    int d = t & 63;
    int r = row[e], c = col[e];
    float nrm = dis[r] * w[e] * dis[c];
    atomicAdd(&agg[(size_t)c * HID + d], nrm * ht[(size_t)r * HID + d]);
}

__global__ void k_bias_relu(float* __restrict__ h, const float* __restrict__ b) {
    int t = blockIdx.x * blockDim.x + threadIdx.x;   // NNODES*HID threads
    float v = h[t] + b[t & 63];
    h[t] = v > 0.0f ? v : 0.0f;
}

// ---------------- head ----------------
__global__ void k_pad_wf(const float* __restrict__ Wf, float* __restrict__ Wfp) {
    int t = blockIdx.x * blockDim.x + threadIdx.x;   // HID*OPAD threads
    if (t < HID * OPAD) {
        int k = t / OPAD, n = t % OPAD;
        Wfp[t] = (n < ODIM) ? Wf[k * ODIM + n] : 0.0f;
    }
}

__global__ void k_log_softmax(const float* __restrict__ logits_pad,
                              const float* __restrict__ bf,
                              float* __restrict__ out) {
    int i = blockIdx.x * blockDim.x + threadIdx.x;   // one node per thread
    if (i >= NNODES) return;
    const float* lr = logits_pad + (size_t)i * OPAD;
    float v[ODIM];
    float mx = -3.402823e38f;
#pragma unroll
    for (int j = 0; j < ODIM; ++j) { v[j] = lr[j] + bf[j]; mx = fmaxf(mx, v[j]); }
    float s = 0.0f;
#pragma unroll
    for (int j = 0; j < ODIM; ++j) s += expf(v[j] - mx);
    float lse = mx + logf(s);
    float* o = out + (size_t)i * ODIM;
#pragma unroll
    for (int j = 0; j < ODIM; ++j) o[j] = v[j] - lse;
}

// ---------------------------------------------------------------------------
extern "C" void kernel_launch(void* const* d_in, const int* in_sizes, int n_in,
                              void* d_out, int out_size, void* d_ws, size_t ws_size,
                              hipStream_t stream) {
    const float* x  = (const float*)d_in[0];
    const float* ew = (const float*)d_in[1];
    const float* W1 = (const float*)d_in[2];
    const float* b1 = (const float*)d_in[3];
    const float* W2 = (const float*)d_in[4];
    const float* b2 = (const float*)d_in[5];
    const float* Wf = (const float*)d_in[6];
    const float* bf = (const float*)d_in[7];
    const int*   ei = (const int*)d_in[8];
    const int* row = ei;
    const int* col = ei + NEDGES;
    float* out = (float*)d_out;

    // workspace layout (floats)
    float* wsf = (float*)d_ws;
    float* dis = wsf;                          // 65536           (deg -> dis)
    float* wfp = wsf + 65536;                  // 64*48 = 3072
    float* ht  = wsf + 131072;                 // 65536*64  (also holds 65536*48 logits)
    float* agg = wsf + 131072 + (size_t)NNODES * HID; // 65536*64

    const int NT  = NNODES * HID;      // 4 194 304 elementwise threads
    const int ET  = NEDGES * HID;      // 67 108 864 scatter threads

    // --- GCN normalization: deg -> dis ---
    k_init_deg   <<<NNODES / 256, 256, 0, stream>>>(dis);
    k_accum_deg  <<<NEDGES / 256, 256, 0, stream>>>(col, ew, dis);
    k_finalize_dis<<<NNODES / 256, 256, 0, stream>>>(dis);
    k_pad_wf     <<<(HID * OPAD + 255) / 256, 256, 0, stream>>>(Wf, wfp);

    // --- layer 1: ht = x @ W1 ; agg = scatter(norm * ht) ; relu(agg + b1) ---
    {
        int tiles = (NNODES / 16) * (HID / 16);   // 16384 waves
        k_gemm_f32<<<dim3(tiles / 4), dim3(32, 4), 0, stream>>>(x, W1, ht, NNODES, INDIM, HID);
    }
    k_self_init   <<<NT / 256, 256, 0, stream>>>(ht, dis, agg);
    k_edge_scatter<<<ET / 256, 256, 0, stream>>>(row, col, ew, dis, ht, agg);
    k_bias_relu   <<<NT / 256, 256, 0, stream>>>(agg, b1);

    // --- layer 2: ht = h1 @ W2 ; agg = scatter ; relu(agg + b2) ---
    {
        int tiles = (NNODES / 16) * (HID / 16);
        k_gemm_f32<<<dim3(tiles / 4), dim3(32, 4), 0, stream>>>(agg, W2, ht, NNODES, HID, HID);
    }
    k_self_init   <<<NT / 256, 256, 0, stream>>>(ht, dis, agg);
    k_edge_scatter<<<ET / 256, 256, 0, stream>>>(row, col, ew, dis, ht, agg);
    k_bias_relu   <<<NT / 256, 256, 0, stream>>>(agg, b2);

    // --- head: logits(padded) = h2 @ Wf_pad ; log_softmax over first 40 ---
    {
        int tiles = (NNODES / 16) * (OPAD / 16);  // 12288 waves
        k_gemm_f32<<<dim3(tiles / 4), dim3(32, 4), 0, stream>>>(agg, wfp, ht, NNODES, HID, OPAD);
    }
    k_log_softmax<<<NNODES / 256, 256, 0, stream>>>(ht, bf, out);
}